// GetInfluences3d_53635551592644
// MI455X (gfx1250) — compile-verified
//
#include <hip/hip_runtime.h>
#include <math.h>

typedef __attribute__((ext_vector_type(2))) float v2f;
typedef __attribute__((ext_vector_type(8))) float v8f;

#define NSTONE 361
#define NTILE  23          // ceil(361/16)
#define NPAD   (NTILE*16)  // 368
#define PBOARD 1024

// d = cosdiff + (equal-stone ? -8 : +4)   (A stone features negated, B positive)
// predicate (different stones AND angdiff<45deg)  <=>  d > 4 + cos45
#define PRED_HI 4.707106781186548f

static_assert(NTILE == 23, "serpentine schedule assumes 23 tiles");

__global__ __launch_bounds__(256)
void influences_wmma_kernel(const float* __restrict__ sda, float* __restrict__ out) {
    __shared__ v2f   fA[NPAD][2];   // A-side: [cos,sin] , [-w*u1, -w*u2]
    __shared__ v2f   fB[NPAD][2];   // B-side: [cos,sin] , [+w*u1, +w*u2]
    __shared__ float gS[NPAD];      // infl_base * stone (0 for padding)
    __shared__ float partial[8];

    const int p    = blockIdx.x;
    const int tid  = threadIdx.x;
    const int lane = tid & 31;      // wave32
    // wave id as a proven-scalar: tile loops lower to SALU control flow,
    // EXEC stays all-ones around every WMMA (ISA requirement).
    const int wid  = __builtin_amdgcn_readfirstlane(tid >> 5);

    const float MAXD = 26.870058f;              // sqrt(19^2+19^2) in f32
    const float W    = 2.8284271247461903f;     // sqrt(8)

    // ---- Stage per-stone features into LDS (padded rows -> 0) ----
    for (int idx = tid; idx < NPAD; idx += 256) {
        float c = 0.f, s = 0.f, u1 = 0.f, u2 = 0.f, g = 0.f;
        if (idx < NSTONE) {
            const float* ptr = sda + ((size_t)p * NSTONE + idx) * 3;
            float stone = ptr[0];
            float dist  = ptr[1];
            float ang   = ptr[2];
            __sincosf(ang * 0.017453292519943295f, &s, &c);
            u1 = W * ((stone == 0.0f) ? 1.0f : -0.5f);       // 120-deg stone phases
            u2 = (W * 0.8660254037844387f) * stone;
            float infl = (MAXD - dist) / MAXD;               // normalize
            if (infl < 0.5f) infl *= 0.5f;                   // distance bias
            g = infl * stone;                                // fold stone weight in
        }
        v2f cs; cs.x = c;   cs.y = s;
        v2f up; up.x = u1;  up.y = u2;
        v2f un; un.x = -u1; un.y = -u2;
        fA[idx][0] = cs; fA[idx][1] = un;
        fB[idx][0] = cs; fB[idx][1] = up;
        gS[idx] = g;
    }
    __syncthreads();

    // A/B 16x4 f32 fragment: lanes 0-15 hold K=0,1 (cos,sin); lanes 16-31 hold K=2,3 (stone)
    const int half = lane >> 4;
    const int l16  = lane & 15;

    // Diagonal-tile row mask (lane-only): row half*8+r counts iff < col l16
    unsigned diagmask = 0;
    #pragma unroll
    for (int r = 0; r < 8; ++r)
        diagmask |= ((half * 8 + r) < l16) ? (1u << r) : 0u;
    const float lowerhalf = (half == 0) ? 1.0f : 0.0f;

    float wsum = 0.0f;

    // Serpentine tile schedule: wave w -> {22-w, 7+w, 6-w}; per-wave WMMA cost 31..38
    #pragma unroll
    for (int k = 0; k < 3; ++k) {
        if (k == 2 && wid == 7) break;             // scalar branch
        const int ti = (k == 0) ? (22 - wid) : ((k == 1) ? (7 + wid) : (6 - wid));

        const int iglob = ti * 16 + l16;           // this lane's target column i
        const v2f b = fB[iglob][half];

        int cnt0 = 0, cnt1 = 0;
        // Strictly-lower tiles: o < i guaranteed -> single f32 compare per element.
        // 2-way unroll with independent accumulators to pipeline WMMAs.
        int to = 0;
        for (; to + 2 <= ti; to += 2) {
            v2f a0 = fA[(to    ) * 16 + l16][half];
            v2f a1 = fA[(to + 1) * 16 + l16][half];
            v8f acc0 = {}, acc1 = {};
            acc0 = __builtin_amdgcn_wmma_f32_16x16x4_f32(
                false, a0, false, b, (short)0, acc0, false, false);
            acc1 = __builtin_amdgcn_wmma_f32_16x16x4_f32(
                false, a1, false, b, (short)0, acc1, false, false);
            #pragma unroll
            for (int r = 0; r < 8; ++r) {
                cnt0 += (acc0[r] > PRED_HI) ? 1 : 0;
                cnt1 += (acc1[r] > PRED_HI) ? 1 : 0;
            }
        }
        if (to < ti) {
            v2f a = fA[to * 16 + l16][half];
            v8f acc = {};
            acc = __builtin_amdgcn_wmma_f32_16x16x4_f32(
                false, a, false, b, (short)0, acc, false, false);
            #pragma unroll
            for (int r = 0; r < 8; ++r)
                cnt0 += (acc[r] > PRED_HI) ? 1 : 0;
        }
        // Diagonal tile: per-element o<i mask (precomputed, lane-only)
        {
            v2f a = fA[ti * 16 + l16][half];
            v8f acc = {};
            acc = __builtin_amdgcn_wmma_f32_16x16x4_f32(
                false, a, false, b, (short)0, acc, false, false);
            #pragma unroll
            for (int r = 0; r < 8; ++r)
                cnt1 += ((acc[r] > PRED_HI) && ((diagmask >> r) & 1u)) ? 1 : 0;
        }

        int cnt = cnt0 + cnt1;
        // merge the two row-halves (column i lives in lane and lane^16)
        cnt += __shfl_xor(cnt, 16, 32);

        // angle bias: infl *= 0.5^count (exact); upper half contributes 0
        wsum += ldexpf(gS[iglob], -cnt) * lowerhalf;
    }

    // single cross-lane reduction after all tiles
    #pragma unroll
    for (int m = 16; m >= 1; m >>= 1) wsum += __shfl_xor(wsum, m, 32);

    if (lane == 0) partial[wid] = wsum;
    __syncthreads();
    if (tid == 0) {
        float total = 0.0f;
        #pragma unroll
        for (int wv = 0; wv < 8; ++wv) total += partial[wv];
        out[p] = total;
    }
}

extern "C" void kernel_launch(void* const* d_in, const int* in_sizes, int n_in,
                              void* d_out, int out_size, void* d_ws, size_t ws_size,
                              hipStream_t stream) {
    const float* sda = (const float*)d_in[0];   // [1024, 361, 3] f32
    float* out = (float*)d_out;                 // [1024] f32
    (void)in_sizes; (void)n_in; (void)out_size; (void)d_ws; (void)ws_size;
    influences_wmma_kernel<<<PBOARD, 256, 0, stream>>>(sda, out);
}